// CausalCompBig_69157563400710
// MI455X (gfx1250) — compile-verified
//
#include <hip/hip_runtime.h>

typedef __attribute__((ext_vector_type(2))) float v2f;
typedef __attribute__((ext_vector_type(8))) float v8f;

#define WMMA_F32X4(a, b, c) \
  __builtin_amdgcn_wmma_f32_16x16x4_f32(false, (a), false, (b), (short)0, (c), false, false)

// Problem constants (reference: B,T,K,D,H,NT = 128,9,8,8,256,8; steps = min(T-1, rollout)=8)
#define BN 128
#define TN 9
#define KN 8
#define DN 8
#define HN 256
#define NTY 8
#define STEPS 8

// ---------------------------------------------------------------------------
// Generic WMMA f32 GEMM: C[M,N] = act(A[M,Kd] @ B[Kd,N] + bias) (+ resid)
// Block: 128 threads (4 waves). Wave w owns cols [blockIdx.x*64 + w*16, +16).
// Grid: (ceil(N/64), M/16). M multiple of 16; Kd multiple of 4, Kd <= 512.
// A tile (16 x Kd) staged in LDS and shared by all 4 waves.
// ---------------------------------------------------------------------------
#define ASTR 516  // padded LDS stride (max Kd 512 + 4)

__global__ __launch_bounds__(128) void gemm_wmma_f32(
    const float* __restrict__ A, const float* __restrict__ B,
    const float* __restrict__ bias, const float* __restrict__ resid,
    float* __restrict__ C, int M, int Kd, int N, int act) {
  __shared__ float As[16 * ASTR];

  const int tid  = threadIdx.x;
  const int wave = tid >> 5;
  const int lane = tid & 31;
  const int half = lane >> 4;
  const int l16  = lane & 15;
  const int m0 = blockIdx.y * 16;
  const int n0 = blockIdx.x * 64 + wave * 16;

  // Cooperative stage of A tile (all threads; single barrier).
  for (int r = 0; r < 16; ++r) {
    const float* Ar = A + (size_t)(m0 + r) * Kd;
    for (int c = tid; c < Kd; c += 128) As[r * ASTR + c] = Ar[c];
  }
  __syncthreads();

  if (n0 >= N) return;  // wave-uniform; no barriers after this point

  const int col = n0 + l16;
  v8f acc = {};

  if (n0 + 16 <= N) {
    // Full tile: completely unguarded hot loop.
    const float* Bp = B + col;
    for (int kb = 0; kb < Kd; kb += 4) {
      const int k = kb + 2 * half;
      v2f a; a.x = As[l16 * ASTR + k]; a.y = As[l16 * ASTR + k + 1];
      v2f b; b.x = Bp[(size_t)k * N];  b.y = Bp[(size_t)(k + 1) * N];
      acc = WMMA_F32X4(a, b, acc);
    }
    const float bv = bias ? bias[col] : 0.0f;
#pragma unroll
    for (int v = 0; v < 8; ++v) {
      const int row = m0 + v + 8 * half;
      float x = acc[v] + bv;
      if (act) x = fmaxf(x, 0.0f);
      if (resid) x += resid[(size_t)row * N + col];
      C[(size_t)row * N + col] = x;
    }
  } else {
    // Partial tile (only the N=8 decoder GEMM): clamped column + mask, branch-free.
    const bool cok = (col < N);
    const int  colc = cok ? col : (N - 1);
    const float msk = cok ? 1.0f : 0.0f;
    const float* Bp = B + colc;
    for (int kb = 0; kb < Kd; kb += 4) {
      const int k = kb + 2 * half;
      v2f a; a.x = As[l16 * ASTR + k];      a.y = As[l16 * ASTR + k + 1];
      v2f b; b.x = Bp[(size_t)k * N] * msk; b.y = Bp[(size_t)(k + 1) * N] * msk;
      acc = WMMA_F32X4(a, b, acc);
    }
    if (cok) {
      const float bv = bias ? bias[col] : 0.0f;
#pragma unroll
      for (int v = 0; v < 8; ++v) {
        const int row = m0 + v + 8 * half;
        float x = acc[v] + bv;
        if (act) x = fmaxf(x, 0.0f);
        if (resid) x += resid[(size_t)row * N + col];
        C[(size_t)row * N + col] = x;
      }
    }
  }
}

// ---------------------------------------------------------------------------
// Fused dynamics message kernel.
// Grid: B*K/2 = 512 WGs; each handles pairs p0..p0+15 = rows (bi0,bi0+1) x j=0..7.
// Computes agg[bi, :] = sum_{j,t} w[b,i,j,t] * (relu(pair @ wm1[t]+bm1[t]) @ wm2[t]+bm2[t])
// ---------------------------------------------------------------------------
#define PS  516   // padded LDS stride for 512-wide pair rows
#define HMS 260   // padded LDS stride for 256-wide hidden rows

__global__ __launch_bounds__(128) void dyn_msg_kernel(
    const float* __restrict__ slots, const float* __restrict__ wbuf,
    const float* __restrict__ wm1, const float* __restrict__ bm1,
    const float* __restrict__ wm2, const float* __restrict__ bm2,
    float* __restrict__ agg) {
  __shared__ float P[16 * PS];
  __shared__ float HM[16 * HMS];

  const int tid  = threadIdx.x;
  const int wave = tid >> 5;
  const int lane = tid & 31;
  const int half = lane >> 4;
  const int l16  = lane & 15;

  const int p0  = blockIdx.x * 16;   // first pair index
  const int bi0 = blockIdx.x * 2;    // first (b,i) row
  const int b   = bi0 >> 3;          // batch index (K=8)

  // Stage pair matrix P[16][512] = [slots[b,i,:], slots[b,j,:]] in LDS.
  for (int idx = tid; idx < 16 * 512; idx += 128) {
    const int r = idx >> 9;
    const int c = idx & 511;
    const int bi = bi0 + (r >> 3);
    const int j  = r & 7;
    const float v = (c < 256) ? slots[(size_t)bi * HN + c]
                              : slots[(size_t)(b * KN + j) * HN + (c & 255)];
    P[r * PS + c] = v;
  }
  __syncthreads();

  float aggAcc[4] = {0.f, 0.f, 0.f, 0.f};

  for (int t = 0; t < NTY; ++t) {
    const float* W1 = wm1 + (size_t)t * 512 * HN;
    const float* B1 = bm1 + t * HN;
    const float* W2 = wm2 + (size_t)t * HN * HN;
    const float* B2 = bm2 + t * HN;

    // Phase 1: HM = relu(P @ W1 + B1).  Wave w -> cols [w*64, w*64+64)
#pragma unroll
    for (int nt = 0; nt < 4; ++nt) {
      const int col = wave * 64 + nt * 16 + l16;
      const float* W1p = W1 + col;
      v8f acc = {};
      for (int kb = 0; kb < 512; kb += 4) {
        const int k = kb + 2 * half;
        v2f a;  a.x = P[l16 * PS + k];        a.y = P[l16 * PS + k + 1];
        v2f bb; bb.x = W1p[(size_t)k * HN];   bb.y = W1p[(size_t)(k + 1) * HN];
        acc = WMMA_F32X4(a, bb, acc);
      }
      const float bv = B1[col];
#pragma unroll
      for (int v = 0; v < 8; ++v) {
        const int row = v + 8 * half;
        HM[row * HMS + col] = fmaxf(acc[v] + bv, 0.0f);
      }
    }
    __syncthreads();

    // Phase 2: M = HM @ W2 + B2, then weighted reduction over pairs (rows).
#pragma unroll
    for (int nt = 0; nt < 4; ++nt) {
      const int col = wave * 64 + nt * 16 + l16;
      const float* W2p = W2 + col;
      v8f acc = {};
      for (int kb = 0; kb < 256; kb += 4) {
        const int k = kb + 2 * half;
        v2f a;  a.x = HM[l16 * HMS + k];      a.y = HM[l16 * HMS + k + 1];
        v2f bb; bb.x = W2p[(size_t)k * HN];   bb.y = W2p[(size_t)(k + 1) * HN];
        acc = WMMA_F32X4(a, bb, acc);
      }
      const float bv = B2[col];
      float part = 0.0f;
#pragma unroll
      for (int v = 0; v < 8; ++v) {
        // D-frag: vgpr v holds row v (lanes 0-15) and row v+8 (lanes 16-31)
        const float wv = wbuf[(size_t)(p0 + v + 8 * half) * NTY + t];
        part += wv * (acc[v] + bv);
      }
      aggAcc[nt] += part;  // lanes 0-15 accumulate bi0, lanes 16-31 bi1
    }
    __syncthreads();
  }

#pragma unroll
  for (int nt = 0; nt < 4; ++nt) {
    const int col = wave * 64 + nt * 16 + l16;
    agg[(size_t)(bi0 + half) * HN + col] = aggAcc[nt];
  }
}

// ---------------------------------------------------------------------------
// Small VALU kernels
// ---------------------------------------------------------------------------

// eps/ets heads + message weights w = et * ep.  One thread per pair row.
__global__ void discover_post_kernel(
    const float* __restrict__ h, const float* __restrict__ gd_we,
    const float* __restrict__ gd_be, const float* __restrict__ gd_wt,
    const float* __restrict__ gd_bt, float* __restrict__ eps_out,
    float* __restrict__ ets_out, float* __restrict__ wbuf, int s) {
  const int r = blockIdx.x * blockDim.x + threadIdx.x;
  if (r >= BN * KN * KN) return;
  const float* hr = h + (size_t)r * HN;
  float e = gd_be[0];
  float tl[NTY];
#pragma unroll
  for (int t = 0; t < NTY; ++t) tl[t] = gd_bt[t];
  for (int c = 0; c < HN; ++c) {
    const float hv = hr[c];
    e += hv * gd_we[c];
#pragma unroll
    for (int t = 0; t < NTY; ++t) tl[t] += hv * gd_wt[c * NTY + t];
  }
  const int j = r & 7, i = (r >> 3) & 7, b = r >> 6;
  float ep = 1.0f / (1.0f + expf(-e));
  if (i == j) ep = 0.0f;
  float mx = tl[0];
#pragma unroll
  for (int t = 1; t < NTY; ++t) mx = fmaxf(mx, tl[t]);
  float sum = 0.0f;
#pragma unroll
  for (int t = 0; t < NTY; ++t) { tl[t] = expf(tl[t] - mx); sum += tl[t]; }
  const float inv = 1.0f / sum;
  eps_out[(size_t)b * (STEPS * 64) + s * 64 + i * 8 + j] = ep;
#pragma unroll
  for (int t = 0; t < NTY; ++t) {
    const float et = tl[t] * inv;
    ets_out[(size_t)b * (STEPS * 512) + s * 512 + (i * 8 + j) * 8 + t] = et;
    wbuf[(size_t)r * NTY + t] = et * ep;
  }
}

// pairbuf[8192][512] = [slots[b,i,:], slots[b,j,:]]
__global__ void pair_build_kernel(const float* __restrict__ slots,
                                  float* __restrict__ pairbuf) {
  const int idx = blockIdx.x * blockDim.x + threadIdx.x;
  if (idx >= BN * KN * KN * 512) return;
  const int r = idx >> 9, c = idx & 511;
  const int b = r >> 6, i = (r >> 3) & 7, j = r & 7;
  pairbuf[idx] = (c < 256) ? slots[(size_t)(b * KN + i) * HN + c]
                           : slots[(size_t)(b * KN + j) * HN + (c & 255)];
}

// ubuf[1024][512] = [slots, agg]
__global__ void concat_u_kernel(const float* __restrict__ slots,
                                const float* __restrict__ agg,
                                float* __restrict__ ubuf) {
  const int idx = blockIdx.x * blockDim.x + threadIdx.x;
  if (idx >= BN * KN * 512) return;
  const int r = idx >> 9, c = idx & 511;
  ubuf[idx] = (c < 256) ? slots[(size_t)r * HN + c] : agg[(size_t)r * HN + (c & 255)];
}

// xbuf[1024][8] = gt_states[:, 0]
__global__ void gather_x0_kernel(const float* __restrict__ gt, float* __restrict__ x) {
  const int idx = blockIdx.x * blockDim.x + threadIdx.x;
  if (idx >= BN * KN * DN) return;
  const int b = idx >> 6, k = (idx >> 3) & 7, d = idx & 7;
  x[idx] = gt[(size_t)b * (TN * KN * DN) + k * DN + d];
}

// target_states = gt_states[:, 1:STEPS+1]
__global__ void target_copy_kernel(const float* __restrict__ gt, float* __restrict__ out) {
  const int idx = blockIdx.x * blockDim.x + threadIdx.x;
  if (idx >= BN * STEPS * KN * DN) return;
  const int d = idx & 7, k = (idx >> 3) & 7, s = (idx >> 6) & 7, b = idx >> 9;
  out[idx] = gt[(size_t)(b * TN + s + 1) * (KN * DN) + k * DN + d];
}

// pred_states[:, s] = predbuf
__global__ void pred_scatter_kernel(const float* __restrict__ pred,
                                    float* __restrict__ out, int s) {
  const int idx = blockIdx.x * blockDim.x + threadIdx.x;
  if (idx >= BN * KN * DN) return;
  const int b = idx >> 6, k = (idx >> 3) & 7, d = idx & 7;
  out[(size_t)b * (STEPS * KN * DN) + s * KN * DN + k * DN + d] = pred[idx];
}

// ---------------------------------------------------------------------------
// Host launcher
// ---------------------------------------------------------------------------
static void launch_gemm(const float* A, const float* B, const float* bias,
                        const float* resid, float* C, int M, int Kd, int N,
                        int act, hipStream_t stream) {
  dim3 grid((N + 63) / 64, M / 16);
  gemm_wmma_f32<<<grid, dim3(128), 0, stream>>>(A, B, bias, resid, C, M, Kd, N, act);
}

extern "C" void kernel_launch(void* const* d_in, const int* in_sizes, int n_in,
                              void* d_out, int out_size, void* d_ws, size_t ws_size,
                              hipStream_t stream) {
  (void)in_sizes; (void)n_in; (void)out_size; (void)ws_size;
  const float* gt      = (const float*)d_in[0];
  const float* enc_w1  = (const float*)d_in[1];
  const float* enc_b1  = (const float*)d_in[2];
  const float* enc_w2  = (const float*)d_in[3];
  const float* enc_b2  = (const float*)d_in[4];
  const float* gd_w1   = (const float*)d_in[5];
  const float* gd_b1   = (const float*)d_in[6];
  const float* gd_we   = (const float*)d_in[7];
  const float* gd_be   = (const float*)d_in[8];
  const float* gd_wt   = (const float*)d_in[9];
  const float* gd_bt   = (const float*)d_in[10];
  const float* dyn_wm1 = (const float*)d_in[11];
  const float* dyn_bm1 = (const float*)d_in[12];
  const float* dyn_wm2 = (const float*)d_in[13];
  const float* dyn_bm2 = (const float*)d_in[14];
  const float* dyn_wu1 = (const float*)d_in[15];
  const float* dyn_bu1 = (const float*)d_in[16];
  const float* dyn_wu2 = (const float*)d_in[17];
  const float* dyn_bu2 = (const float*)d_in[18];
  const float* dec_w1  = (const float*)d_in[19];
  const float* dec_b1  = (const float*)d_in[20];
  const float* dec_w2  = (const float*)d_in[21];
  const float* dec_b2  = (const float*)d_in[22];
  // d_in[23] = rollout_steps (device scalar); steps = min(T-1, 8) = 8, fixed.

  float* out_pred = (float*)d_out;                       // (B, 8, K, D)
  float* out_tgt  = out_pred + BN * STEPS * KN * DN;     // (B, 8, K, D)
  float* out_eps  = out_tgt  + BN * STEPS * KN * DN;     // (B, 8, K, K)
  float* out_ets  = out_eps  + BN * STEPS * KN * KN;     // (B, 8, K, K, NT)

  // Workspace layout (floats)
  float* ws = (float*)d_ws;
  float* slots   = ws;                       ws += BN * KN * HN;          // 262144
  float* pairbuf = ws;                       ws += BN * KN * KN * 512;    // 4194304
  float* hbuf    = ws;                       ws += BN * KN * KN * HN;     // 2097152
  float* wbuf    = ws;                       ws += BN * KN * KN * NTY;    // 65536
  float* aggbuf  = ws;                       ws += BN * KN * HN;          // 262144
  float* ubuf    = ws;                       ws += BN * KN * 512;         // 524288
  float* hidbuf  = ws;                       ws += BN * KN * HN;          // 262144
  float* predbuf = ws;                       ws += BN * KN * DN;          // 8192
  float* xbuf    = ws;                       ws += BN * KN * DN;          // 8192

  const int MSL = BN * KN;            // 1024 slot rows
  const int MPR = BN * KN * KN;       // 8192 pair rows

  // target_states
  target_copy_kernel<<<(BN * STEPS * KN * DN + 255) / 256, 256, 0, stream>>>(gt, out_tgt);

  // slots = enc(gt[:,0])
  gather_x0_kernel<<<(MSL * DN + 255) / 256, 256, 0, stream>>>(gt, xbuf);
  launch_gemm(xbuf, enc_w1, enc_b1, nullptr, hidbuf, MSL, DN, HN, 1, stream);
  launch_gemm(hidbuf, enc_w2, enc_b2, nullptr, slots, MSL, HN, HN, 0, stream);

  for (int s = 0; s < STEPS; ++s) {
    // discover
    pair_build_kernel<<<(MPR * 512 + 255) / 256, 256, 0, stream>>>(slots, pairbuf);
    launch_gemm(pairbuf, gd_w1, gd_b1, nullptr, hbuf, MPR, 512, HN, 1, stream);
    discover_post_kernel<<<(MPR + 255) / 256, 256, 0, stream>>>(
        hbuf, gd_we, gd_be, gd_wt, gd_bt, out_eps, out_ets, wbuf, s);

    // dynamics (2 iterations, fixed w)
    for (int it = 0; it < 2; ++it) {
      dyn_msg_kernel<<<dim3(MSL / 2), dim3(128), 0, stream>>>(
          slots, wbuf, dyn_wm1, dyn_bm1, dyn_wm2, dyn_bm2, aggbuf);
      concat_u_kernel<<<(MSL * 512 + 255) / 256, 256, 0, stream>>>(slots, aggbuf, ubuf);
      launch_gemm(ubuf, dyn_wu1, dyn_bu1, nullptr, hidbuf, MSL, 512, HN, 1, stream);
      launch_gemm(hidbuf, dyn_wu2, dyn_bu2, slots, slots, MSL, HN, HN, 0, stream); // residual
    }

    // decode -> pred
    launch_gemm(slots, dec_w1, dec_b1, nullptr, hidbuf, MSL, HN, HN, 1, stream);
    launch_gemm(hidbuf, dec_w2, dec_b2, nullptr, predbuf, MSL, HN, DN, 0, stream);
    pred_scatter_kernel<<<(MSL * DN + 255) / 256, 256, 0, stream>>>(predbuf, out_pred, s);

    // re-encode pred -> slots
    launch_gemm(predbuf, enc_w1, enc_b1, nullptr, hidbuf, MSL, DN, HN, 1, stream);
    launch_gemm(hidbuf, enc_w2, enc_b2, nullptr, slots, MSL, HN, HN, 0, stream);
  }
}